// GCNLinkPredictor_55490977465137
// MI455X (gfx1250) — compile-verified
//
#include <hip/hip_runtime.h>

typedef float v2f __attribute__((ext_vector_type(2)));
typedef float v8f __attribute__((ext_vector_type(8)));

#define NOUT 128

// ---------------------------------------------------------------------------
// C[M,128] = A[M,K] @ W[K,128], fp32 WMMA 16x16x4.
// Block = 256 threads (8 waves). Each wave computes a 16-row x 128-col strip.
// W is staged into LDS with CDNA5 async global->LDS copies (ASYNCcnt),
// double-buffered in K-panels of 32 so panel p+1 streams behind the WMMAs.
// ---------------------------------------------------------------------------
__global__ __launch_bounds__(256) void wmma_gemm_f32_n128(
    const float* __restrict__ A, const float* __restrict__ W,
    float* __restrict__ C, int M, int K)
{
    const int KP = 32;
    __shared__ float sW[2][KP][NOUT + 4];   // +4 pad to stagger banks (~34 KB)

    const int lane  = threadIdx.x & 31;
    const int wave  = threadIdx.x >> 5;
    const int row0  = blockIdx.x * 128 + wave * 16;
    const int rowA  = row0 + (lane & 15);
    const int khalf = (lane >> 4) << 1;          // 0 for lanes 0-15, 2 for 16-31

    v8f acc[8] = {};                              // 8 tiles of 16x16 f32 accum

    const int np = K / KP;                        // K is 256 or 128

    // async stage of one 32x128 W panel into sW[buf]:
    // 1024 float4s / 256 threads -> exactly 4 async b128 copies per thread
    auto stage = [&](int k0, int buf) {
        for (int i = threadIdx.x * 4; i < KP * NOUT; i += 256 * 4) {
            const int kk = i >> 7;               // /128
            const int nn = i & (NOUT - 1);
            unsigned lds = (unsigned)(size_t)&sW[buf][kk][nn];
            unsigned long long ga =
                (unsigned long long)(W + (size_t)(k0 + kk) * NOUT + nn);
            asm volatile("global_load_async_to_lds_b128 %0, %1, off"
                         :: "v"(lds), "v"(ga) : "memory");
        }
    };

    stage(0, 0);
    for (int p = 0; p < np; ++p) {
        const int buf = p & 1;
        __syncthreads();                         // all waves done reading sW[buf^1]
        if (p + 1 < np) {
            stage((p + 1) * KP, buf ^ 1);        // prefetch next panel
            // async loads retire in order: <=4 outstanding => panel p landed
            asm volatile("s_wait_asynccnt 0x4" ::: "memory");
        } else {
            asm volatile("s_wait_asynccnt 0x0" ::: "memory");
        }
        __syncthreads();                         // panel p resident for all waves

        #pragma unroll
        for (int kk = 0; kk < KP; kk += 4) {
            // A fragment: 16x4 f32 (ISA layout: lanes 0-15 hold K=0,1; 16-31 hold K=2,3)
            v2f a;
            if (rowA < M) {
                const float2 av = *reinterpret_cast<const float2*>(
                    A + (size_t)rowA * K + (p * KP + kk + khalf));
                a[0] = av.x; a[1] = av.y;
            } else {
                a[0] = 0.f; a[1] = 0.f;
            }
            const int kb = kk + khalf;
            #pragma unroll
            for (int t = 0; t < 8; ++t) {
                const int n = t * 16 + (lane & 15);
                v2f b;
                b[0] = sW[buf][kb][n];
                b[1] = sW[buf][kb + 1][n];
                acc[t] = __builtin_amdgcn_wmma_f32_16x16x4_f32(
                    false, a, false, b, (short)0, acc[t], false, false);
            }
        }
    }

    // C/D layout: VGPR v -> row v (lanes 0-15) / row v+8 (lanes 16-31), N = lane%16
    const int cn    = lane & 15;
    const int rbase = row0 + ((lane >> 4) << 3);
    #pragma unroll
    for (int t = 0; t < 8; ++t) {
        #pragma unroll
        for (int v = 0; v < 8; ++v) {
            const int r = rbase + v;
            if (r < M) C[(size_t)r * NOUT + t * 16 + cn] = acc[t][v];
        }
    }
}

// ---------------------------------------------------------------------------
// Degree / normalization kernels (deg = 1 self-loop + count of dst)
// ---------------------------------------------------------------------------
__global__ void deg_init(float* __restrict__ deg, int M)
{
    int i = blockIdx.x * blockDim.x + threadIdx.x;
    if (i < M) deg[i] = 1.0f;
}

__global__ void deg_count(const long long* __restrict__ dst, int E,
                          float* __restrict__ deg)
{
    int e = blockIdx.x * blockDim.x + threadIdx.x;
    if (e < E) atomicAdd(&deg[dst[e]], 1.0f);
}

__global__ void deg_rsqrt(float* __restrict__ deg, int M)
{
    int i = blockIdx.x * blockDim.x + threadIdx.x;
    if (i < M) deg[i] = rsqrtf(deg[i]);     // deg >= 1 always
}

// ---------------------------------------------------------------------------
// out[i,:] = h[i,:] * dinv[i]^2   (self-loop contribution, initializes out)
// one thread per float4 quad
// ---------------------------------------------------------------------------
__global__ void agg_self(const float* __restrict__ h,
                         const float* __restrict__ dinv,
                         float* __restrict__ out, int M)
{
    long long idx = (long long)blockIdx.x * blockDim.x + threadIdx.x;
    int i = (int)(idx >> 5);
    if (i >= M) return;
    int q = (int)(idx & 31);
    float w = dinv[i] * dinv[i];
    const float4 hv = *reinterpret_cast<const float4*>(h + (size_t)i * NOUT + q * 4);
    float4 ov;
    ov.x = hv.x * w; ov.y = hv.y * w; ov.z = hv.z * w; ov.w = hv.w * w;
    *reinterpret_cast<float4*>(out + (size_t)i * NOUT + q * 4) = ov;
}

// ---------------------------------------------------------------------------
// Edge scatter: out[d,:] += h[s,:] * dinv[s]*dinv[d]
// 32 threads (one wave) per edge, float4 per thread -> coalesced 512B rows
// (h and out are L2-resident at 51 MB each; atomics resolve in L2)
// ---------------------------------------------------------------------------
__global__ void agg_scatter(const float* __restrict__ h,
                            const float* __restrict__ dinv,
                            const long long* __restrict__ src,
                            const long long* __restrict__ dst,
                            int E, float* __restrict__ out)
{
    long long idx = (long long)blockIdx.x * blockDim.x + threadIdx.x;
    int e = (int)(idx >> 5);
    if (e >= E) return;
    int q = (int)(idx & 31);
    long long s = src[e];
    long long d = dst[e];
    float w = dinv[s] * dinv[d];
    const float4 hv = *reinterpret_cast<const float4*>(h + (size_t)s * NOUT + q * 4);
    float* op = out + (size_t)d * NOUT + q * 4;
    atomicAdd(op + 0, hv.x * w);
    atomicAdd(op + 1, hv.y * w);
    atomicAdd(op + 2, hv.z * w);
    atomicAdd(op + 3, hv.w * w);
}

// ---------------------------------------------------------------------------
// h[i,f] = (h[i,f] + bias[f])  [+ ReLU]
// ---------------------------------------------------------------------------
__global__ void bias_act(float* __restrict__ h, const float* __restrict__ bias,
                         int M, int do_relu)
{
    long long idx = (long long)blockIdx.x * blockDim.x + threadIdx.x;
    if (idx >= (long long)M * NOUT) return;
    int f = (int)(idx & (NOUT - 1));
    float v = h[idx] + bias[f];
    if (do_relu) v = fmaxf(v, 0.0f);
    h[idx] = v;
}

// ---------------------------------------------------------------------------
// decoder: out[e] = dot(z[ls[e],:], z[ld[e],:]) over 128 dims
// one wave per edge, float4 per lane, shfl reduction (wave32)
// ---------------------------------------------------------------------------
__global__ __launch_bounds__(256) void decode_dot(
    const float* __restrict__ z,
    const long long* __restrict__ ls, const long long* __restrict__ ld,
    int EL, float* __restrict__ out)
{
    long long idx = (long long)blockIdx.x * blockDim.x + threadIdx.x;
    int e    = (int)(idx >> 5);
    int lane = (int)(idx & 31);
    if (e >= EL) return;
    long long a = ls[e];
    long long b = ld[e];
    const float4 za = *reinterpret_cast<const float4*>(z + (size_t)a * NOUT + lane * 4);
    const float4 zb = *reinterpret_cast<const float4*>(z + (size_t)b * NOUT + lane * 4);
    float s = za.x * zb.x + za.y * zb.y + za.z * zb.z + za.w * zb.w;
    #pragma unroll
    for (int off = 16; off > 0; off >>= 1)
        s += __shfl_down(s, off, 32);
    if (lane == 0) out[e] = s;
}

// ---------------------------------------------------------------------------
extern "C" void kernel_launch(void* const* d_in, const int* in_sizes, int n_in,
                              void* d_out, int out_size, void* d_ws, size_t ws_size,
                              hipStream_t stream)
{
    (void)n_in; (void)out_size; (void)ws_size;

    const float*     x   = (const float*)d_in[0];
    const float*     W1  = (const float*)d_in[1];
    const float*     b1  = (const float*)d_in[2];
    const float*     W2  = (const float*)d_in[3];
    const float*     b2  = (const float*)d_in[4];
    const long long* ei  = (const long long*)d_in[5];   // int64 [2, E]
    const long long* eli = (const long long*)d_in[6];   // int64 [2, EL]

    const int M  = in_sizes[0] / 256;   // 100000 nodes
    const int E  = in_sizes[5] / 2;     // 1.6M edges
    const int EL = in_sizes[6] / 2;     // 500K label edges

    const long long* src = ei;
    const long long* dst = ei + E;
    const long long* ls  = eli;
    const long long* ld  = eli + EL;

    // workspace layout: dinv | bufA (M*128) | bufB (M*128)
    float* dinv = (float*)d_ws;
    float* bufA = dinv + (((size_t)M + 255) & ~(size_t)255);
    float* bufB = bufA + (size_t)M * NOUT;
    float* out  = (float*)d_out;

    const unsigned TB = 256;
    const unsigned gM    = (unsigned)((M + TB - 1) / TB);
    const unsigned gE    = (unsigned)((E + TB - 1) / TB);
    const unsigned gMq   = (unsigned)(((long long)M * 32 + TB - 1) / TB);
    const unsigned gEq   = (unsigned)(((long long)E * 32 + TB - 1) / TB);
    const unsigned gMf   = (unsigned)(((long long)M * NOUT + TB - 1) / TB);
    const unsigned gELq  = (unsigned)(((long long)EL * 32 + TB - 1) / TB);
    const unsigned gGemm = (unsigned)((M + 127) / 128);

    // symmetric normalization factors
    deg_init  <<<gM, TB, 0, stream>>>(dinv, M);
    deg_count <<<gE, TB, 0, stream>>>(dst, E, dinv);
    deg_rsqrt <<<gM, TB, 0, stream>>>(dinv, M);

    // layer 1: h = relu(Agg(x @ W1) + b1)
    wmma_gemm_f32_n128<<<gGemm, TB, 0, stream>>>(x, W1, bufA, M, 256);
    agg_self   <<<gMq, TB, 0, stream>>>(bufA, dinv, bufB, M);
    agg_scatter<<<gEq, TB, 0, stream>>>(bufA, dinv, src, dst, E, bufB);
    bias_act   <<<gMf, TB, 0, stream>>>(bufB, b1, M, 1);

    // layer 2: z = Agg(h @ W2) + b2
    wmma_gemm_f32_n128<<<gGemm, TB, 0, stream>>>(bufB, W2, bufA, M, 128);
    agg_self   <<<gMq, TB, 0, stream>>>(bufA, dinv, bufB, M);
    agg_scatter<<<gEq, TB, 0, stream>>>(bufA, dinv, src, dst, E, bufB);
    bias_act   <<<gMf, TB, 0, stream>>>(bufB, b2, M, 0);

    // decoder
    decode_dot<<<gELq, TB, 0, stream>>>(bufB, ls, ld, EL, out);
}